// GNNEncoder_18820546691488
// MI455X (gfx1250) — compile-verified
//
#include <hip/hip_runtime.h>

typedef __attribute__((ext_vector_type(2))) float v2f;
typedef __attribute__((ext_vector_type(8))) float v8f;

#define WAVES_PER_BLOCK 4
#define LN_EPS 1e-5f

__device__ __forceinline__ float silu(float v) {
  return v / (1.0f + __expf(-v));
}

// Repack weight W (K x 128, row-major) into WMMA-B fragment order:
//   packed[(((nt*(K/4) + ks)*32 + L)*2 + j] = W[ks*4 + (L>>4)*2 + j][nt*16 + (L&15)]
// so each lane's B fragment is one coalesced 8-byte load.
__global__ void repack_w_kernel(const float* __restrict__ w, float* __restrict__ packed,
                                int K) {
  const int gid = blockIdx.x * blockDim.x + threadIdx.x;
  if (gid >= K * 128) return;
  const int j    = gid & 1;
  const int L    = (gid >> 1) & 31;
  const int rest = gid >> 6;
  const int ksteps = K >> 2;
  const int ks = rest % ksteps;
  const int nt = rest / ksteps;
  const int k = ks * 4 + ((L >> 4) * 2) + j;
  const int n = nt * 16 + (L & 15);
  packed[gid] = w[k * 128 + n];
}

// Fused: out = LayerNorm(SiLU(relu(in @ w0 + b0) @ w1 + b1))
// One 16-node tile per wave; A tile + relu intermediate staged in LDS
// (row stride 132 floats -> conflict-free fragment loads); weights come
// pre-packed in fragment order (one b64 load per WMMA).
template<int DIN>
__global__ __launch_bounds__(128) void mlp_silu_ln_kernel(
    const float* __restrict__ in, int n_nodes,
    const float* __restrict__ w0p, const float* __restrict__ b0,
    const float* __restrict__ w1p, const float* __restrict__ b1,
    float* __restrict__ out)
{
  __shared__ float smem[WAVES_PER_BLOCK][16 * 132];
  const int lane = threadIdx.x & 31;
  const int wv   = threadIdx.x >> 5;
  float* sm = &smem[wv][0];
  const int tile = blockIdx.x * WAVES_PER_BLOCK + wv;
  const int row0 = tile * 16;

  // stage 16 x DIN input tile into LDS (clamped rows; keeps EXEC full)
  for (int idx = lane * 4; idx < 16 * DIN; idx += 32 * 4) {
    const int r = idx / DIN;
    const int c = idx - r * DIN;
    const int gr = min(row0 + r, n_nodes - 1);
    const float4 v = *(const float4*)(in + (size_t)gr * DIN + c);
    *(float4*)(sm + r * 132 + c) = v;
  }
  __syncthreads();

  const int m    = lane & 15;        // A row / B,C column handled by this lane
  const int koff = (lane >> 4) * 2;  // K sub-offset for A fragments
  const int hi   = (lane >> 4) * 8;  // C/D row offset for this half-wave

  v8f acc[8];

  // ---- GEMM1: t = relu(A(16xDIN) @ w0(DINx128) + b0) ----
  for (int nt = 0; nt < 8; ++nt) {
    const float bias = b0[nt * 16 + m];
    v8f c8;
#pragma unroll
    for (int i = 0; i < 8; ++i) c8[i] = bias;
#pragma unroll
    for (int ks = 0; ks < DIN / 4; ++ks) {
      v2f a;
      a.x = sm[m * 132 + ks * 4 + koff];
      a.y = sm[m * 132 + ks * 4 + koff + 1];
      const v2f b = *(const v2f*)(w0p + (((nt * (DIN / 4) + ks) * 32) + lane) * 2);
      c8 = __builtin_amdgcn_wmma_f32_16x16x4_f32(false, a, false, b, (short)0, c8,
                                                 false, false);
    }
#pragma unroll
    for (int i = 0; i < 8; ++i) c8[i] = fmaxf(c8[i], 0.0f);
    acc[nt] = c8;
  }
  __syncthreads();

  // write t over the LDS tile (becomes A for GEMM2)
#pragma unroll
  for (int nt = 0; nt < 8; ++nt)
#pragma unroll
    for (int v = 0; v < 8; ++v)
      sm[(v + hi) * 132 + nt * 16 + m] = acc[nt][v];
  __syncthreads();

  // ---- GEMM2: z = t(16x128) @ w1(128x128) + b1 ----
  for (int nt = 0; nt < 8; ++nt) {
    const float bias = b1[nt * 16 + m];
    v8f c8;
#pragma unroll
    for (int i = 0; i < 8; ++i) c8[i] = bias;
#pragma unroll
    for (int ks = 0; ks < 32; ++ks) {
      v2f a;
      a.x = sm[m * 132 + ks * 4 + koff];
      a.y = sm[m * 132 + ks * 4 + koff + 1];
      const v2f b = *(const v2f*)(w1p + (((nt * 32 + ks) * 32) + lane) * 2);
      c8 = __builtin_amdgcn_wmma_f32_16x16x4_f32(false, a, false, b, (short)0, c8,
                                                 false, false);
    }
    acc[nt] = c8;
  }

  // ---- SiLU then LayerNorm over each row's 128 features ----
  float s[8], q[8];
#pragma unroll
  for (int v = 0; v < 8; ++v) { s[v] = 0.0f; q[v] = 0.0f; }
#pragma unroll
  for (int nt = 0; nt < 8; ++nt)
#pragma unroll
    for (int v = 0; v < 8; ++v) {
      float z = acc[nt][v];
      z = silu(z);
      acc[nt][v] = z;
      s[v] += z;
      q[v] += z * z;
    }
  // reduce across the 16 lanes of each half-wave (rows live within halves)
#pragma unroll
  for (int off = 1; off < 16; off <<= 1)
#pragma unroll
    for (int v = 0; v < 8; ++v) {
      s[v] += __shfl_xor(s[v], off, 32);
      q[v] += __shfl_xor(q[v], off, 32);
    }
  float mu[8], inv[8];
#pragma unroll
  for (int v = 0; v < 8; ++v) {
    mu[v] = s[v] * (1.0f / 128.0f);
    const float var = q[v] * (1.0f / 128.0f) - mu[v] * mu[v];
    inv[v] = rsqrtf(var + LN_EPS);
  }

  // write normalized rows
#pragma unroll
  for (int nt = 0; nt < 8; ++nt)
#pragma unroll
    for (int v = 0; v < 8; ++v) {
      const int gr = row0 + v + hi;
      if (gr < n_nodes)
        out[(size_t)gr * 128 + nt * 16 + m] = (acc[nt][v] - mu[v]) * inv[v];
    }
}

__global__ void init_zero_kernel(float* __restrict__ agg, float* __restrict__ cnt,
                                 int n_nodes) {
  const size_t i = (size_t)blockIdx.x * blockDim.x + threadIdx.x;
  if (i < (size_t)n_nodes * 128) agg[i] = 0.0f;
  if (i < (size_t)n_nodes) cnt[i] = 0.0f;
}

__global__ void degree_kernel(const int* __restrict__ tgt, float* __restrict__ cnt,
                              int n_edges) {
  const int e = blockIdx.x * blockDim.x + threadIdx.x;
  if (e < n_edges) atomicAdd(&cnt[tgt[e]], 1.0f);
}

// one wave per edge: 4 floats per lane; gather/atomics hit L2 (working set < 192MB)
__global__ void scatter_kernel(const float* __restrict__ z, const int* __restrict__ src,
                               const int* __restrict__ tgt, float* __restrict__ agg,
                               int n_edges) {
  const int gid  = blockIdx.x * blockDim.x + threadIdx.x;
  const int e    = gid >> 5;
  const int lane = gid & 31;
  if (e >= n_edges) return;
  const int sN = src[e];
  const int tN = tgt[e];
  const float4 v = *(const float4*)(z + (size_t)sN * 128 + lane * 4);
  float* a = agg + (size_t)tN * 128 + lane * 4;
  atomicAdd(a + 0, v.x);
  atomicAdd(a + 1, v.y);
  atomicAdd(a + 2, v.z);
  atomicAdd(a + 3, v.w);
}

// h += agg / max(cnt,1); also re-zeroes agg for the next layer / next call
__global__ void combine_kernel(float* __restrict__ agg, const float* __restrict__ cnt,
                               float* __restrict__ h, int n_nodes) {
  const size_t i = (size_t)blockIdx.x * blockDim.x + threadIdx.x;
  if (i >= (size_t)n_nodes * 128) return;
  const float c = cnt[i >> 7];
  h[i] += agg[i] / fmaxf(c, 1.0f);
  agg[i] = 0.0f;
}

extern "C" void kernel_launch(void* const* d_in, const int* in_sizes, int n_in,
                              void* d_out, int out_size, void* d_ws, size_t ws_size,
                              hipStream_t stream) {
  const float* x     = (const float*)d_in[0];
  const int*   eidx  = (const int*)d_in[1];
  const float* ie_w0 = (const float*)d_in[2];
  const float* ie_b0 = (const float*)d_in[3];
  const float* ie_w1 = (const float*)d_in[4];
  const float* ie_b1 = (const float*)d_in[5];
  const float* lw0   = (const float*)d_in[6];
  const float* lb0   = (const float*)d_in[7];
  const float* lw1   = (const float*)d_in[8];
  const float* lb1   = (const float*)d_in[9];

  const int N = in_sizes[0] / 64;   // D_IN = 64
  const int E = in_sizes[1] / 2;    // edge_index is (2, E)
  const int* src = eidx;
  const int* tgt = eidx + E;

  float* z   = (float*)d_ws;                 // N*128
  float* agg = z + (size_t)N * 128;          // N*128
  float* cnt = agg + (size_t)N * 128;        // N
  // packed weights (16-float aligned)
  size_t off = (size_t)N * 257;
  off = (off + 15) & ~(size_t)15;
  float* pw_ie0 = (float*)d_ws + off;            // 64*128
  float* pw_ie1 = pw_ie0 + 64 * 128;             // 128*128
  float* pw_l0  = pw_ie1 + 128 * 128;            // 3 * 128*128
  float* pw_l1  = pw_l0 + 3 * 128 * 128;         // 3 * 128*128
  float* h = (float*)d_out;                      // N*128 (final output)

  const dim3 blk(256);
  const dim3 grd_feat((N * 128 + 255) / 256);

  // weight repack (cheap; redone every call -> no cross-call state)
  repack_w_kernel<<<(64 * 128 + 255) / 256, blk, 0, stream>>>(ie_w0, pw_ie0, 64);
  repack_w_kernel<<<(128 * 128 + 255) / 256, blk, 0, stream>>>(ie_w1, pw_ie1, 128);
  for (int l = 0; l < 3; ++l) {
    repack_w_kernel<<<(128 * 128 + 255) / 256, blk, 0, stream>>>(
        lw0 + (size_t)l * 128 * 128, pw_l0 + (size_t)l * 128 * 128, 128);
    repack_w_kernel<<<(128 * 128 + 255) / 256, blk, 0, stream>>>(
        lw1 + (size_t)l * 128 * 128, pw_l1 + (size_t)l * 128 * 128, 128);
  }

  init_zero_kernel<<<grd_feat, blk, 0, stream>>>(agg, cnt, N);
  degree_kernel<<<(E + 255) / 256, blk, 0, stream>>>(tgt, cnt, E);

  const int tiles = (N + 15) / 16;
  const int mlp_blocks = (tiles + WAVES_PER_BLOCK - 1) / WAVES_PER_BLOCK;
  mlp_silu_ln_kernel<64><<<mlp_blocks, 128, 0, stream>>>(x, N, pw_ie0, ie_b0,
                                                         pw_ie1, ie_b1, h);

  const int scat_blocks = (int)(((size_t)E * 32 + 255) / 256);
  for (int l = 0; l < 3; ++l) {
    mlp_silu_ln_kernel<128><<<mlp_blocks, 128, 0, stream>>>(
        h, N,
        pw_l0 + (size_t)l * 128 * 128, lb0 + (size_t)l * 128,
        pw_l1 + (size_t)l * 128 * 128, lb1 + (size_t)l * 128, z);
    scatter_kernel<<<scat_blocks, blk, 0, stream>>>(z, src, tgt, agg, E);
    combine_kernel<<<grd_feat, blk, 0, stream>>>(agg, cnt, h, N);
  }
}